// PSSM_Projector_4458176053662
// MI455X (gfx1250) — compile-verified
//
#include <hip/hip_runtime.h>
#include <hip/hip_bf16.h>

typedef __attribute__((ext_vector_type(2))) float v2f;
typedef __attribute__((ext_vector_type(8))) float v8f;

#define B_  8
#define S_  2048
#define K_  21
#define L_  16
#define CM_ 256
#define WPB 8   // waves per block (256 threads = 8 wave32)

// out[b,l,s,c] = sum_k x[b,s,k] * W[l,c,k] + bias[l,c]
// One wave computes a 16(s) x 16(c) tile via 6x V_WMMA_F32_16X16X4_F32 (K 21->24 pad),
// iterating over the 16 c-tiles of CM=256 while reusing its A fragment.
// K-tail (k=20) handled with value-selects (v_cndmask), not exec divergence.
__global__ __launch_bounds__(256) void pssm_proj_wmma_f32(
    const float* __restrict__ x,     // [B, S, K]
    const float* __restrict__ W,     // [L, CM, K]
    const float* __restrict__ bias,  // [L, CM]
    float* __restrict__ out)         // [B, L, S, CM]
{
    const int lane = threadIdx.x & 31;
    const int wave = threadIdx.x >> 5;
    const int hl   = lane & 15;   // 0..15 : M (A) or N (B/C/D) index
    const int hi   = lane >> 4;   // 0/1   : selects K pair (A/B) or M+8 (C/D)

    const int ST   = S_ / 16;                      // 128 s-tiles
    const int tile = blockIdx.x * WPB + wave;      // exact: grid covers B*L*ST
    const int st   = tile % ST;
    const int bl   = tile / ST;
    const int l    = bl % L_;
    const int bb   = bl / L_;

    const int s0 = st * 16;
    const float* xrow = x + ((size_t)(bb * S_ + s0) + hl) * K_;   // this lane's A row
    const float* Wl   = W + (size_t)l * CM_ * K_;                 // [256, 21]
    const float* bv_p = bias + l * CM_;
    float* outp = out + (((size_t)(bb * L_ + l)) * S_ + s0) * CM_;

    // ---- A fragments (x tile 16x21, padded to 16x24): 6 k-steps of 4 ----
    // ISA 32-bit A 16x4 layout: VGPR0 = K=4kk+2*hi, VGPR1 = +1, rows = lane%16.
    v2f a[6];
#pragma unroll
    for (int kk = 0; kk < 5; ++kk) {               // kk=0..4 fully in-bounds (max k=19)
        const int k0 = kk * 4 + hi * 2;
        a[kk].x = xrow[k0 + 0];
        a[kk].y = xrow[k0 + 1];
    }
    {   // kk=5 tail: k={20,21} (hi=0) / {22,23} (hi=1); only k=20 is real
        const float x20 = xrow[20];                // valid address for all lanes
        a[5].x = hi ? 0.0f : x20;                  // v_cndmask, no divergence
        a[5].y = 0.0f;
    }

    // ---- iterate over 16 column tiles of CM=256 ----
#pragma unroll 1
    for (int ct = 0; ct < CM_ / 16; ++ct) {
        const int c0 = ct * 16;
        const float* wrow = Wl + (size_t)(c0 + hl) * K_;   // B col n = lane%16

        const float bv = bv_p[c0 + hl];                    // bias[l, c0+hl]
        v8f acc = {bv, bv, bv, bv, bv, bv, bv, bv};

        v2f bf[6];
#pragma unroll
        for (int kk = 0; kk < 5; ++kk) {           // uniform b64 fragment loads
            const int k0 = kk * 4 + hi * 2;
            bf[kk].x = wrow[k0 + 0];
            bf[kk].y = wrow[k0 + 1];
        }
        {   // tail: single b32 + value-select, no exec branch
            const float w20 = wrow[20];            // valid address for all lanes
            bf[5].x = hi ? 0.0f : w20;
            bf[5].y = 0.0f;
        }

#pragma unroll
        for (int kk = 0; kk < 6; ++kk) {
            acc = __builtin_amdgcn_wmma_f32_16x16x4_f32(
                false, a[kk], false, bf[kk], (short)0, acc,
                /*reuse_a=*/false, /*reuse_b=*/false);
        }

        // ---- store: C/D VGPR r -> row (r + 8*hi), col c0+hl; NT to bypass L2 ----
        float* orow = outp + (size_t)(8 * hi) * CM_ + (c0 + hl);
#pragma unroll
        for (int r = 0; r < 8; ++r) {
            __builtin_nontemporal_store(acc[r], orow + (size_t)r * CM_);
        }
    }
}

extern "C" void kernel_launch(void* const* d_in, const int* in_sizes, int n_in,
                              void* d_out, int out_size, void* d_ws, size_t ws_size,
                              hipStream_t stream) {
    (void)in_sizes; (void)n_in; (void)d_ws; (void)ws_size; (void)out_size;
    const float* x    = (const float*)d_in[0];  // [8, 2048, 21]
    const float* W    = (const float*)d_in[1];  // [16, 256, 21]
    const float* bias = (const float*)d_in[2];  // [16, 256]
    float* out        = (float*)d_out;          // [8, 16, 2048, 256]

    const int total_waves = B_ * L_ * (S_ / 16);   // 16384
    const int blocks = total_waves / WPB;          // 2048, exact
    pssm_proj_wmma_f32<<<blocks, 32 * WPB, 0, stream>>>(x, W, bias, out);
}